// LabelPropagation_6975026888995
// MI455X (gfx1250) — compile-verified
//
#include <hip/hip_runtime.h>
#include <stdint.h>

// Label propagation, DAD normalization, 3 layers, alpha = 0.9.
//   y0 = mask * labels ; last = 0.1 * y0 ; norm = deg^-1/2
//   per layer: y = clip(last + 0.9 * norm_d * sum_{src->d} (norm_s * y_s), 0, 1)
//
// MI455X strategy: feature matrix (25.6 MB) is L2-resident (192 MB L2), so the
// per-layer 819 MB gather is L2 bandwidth, not HBM.  Avoid the 205M/layer f32
// atomics of a scatter formulation (L2 atomic-pipe bound) by building CSR once
// (degree -> hierarchical wave32 scan -> ticket fill) and doing a register-
// accumulated gather per destination node, fused with the norm/clip update.
// Edge-index streams are staged to LDS with CDNA5 async loads (ASYNCcnt).

#define ALPHA_F 0.9f
#define LAST_F  0.1f   // (1 - alpha)

#ifndef __has_builtin
#define __has_builtin(x) 0
#endif

#if defined(__HIP_DEVICE_COMPILE__) && __has_builtin(__builtin_amdgcn_global_load_async_to_lds_b32)
#define HAVE_ASYNC_LDS 1
#else
#define HAVE_ASYNC_LDS 0
#endif

typedef __attribute__((address_space(1))) int g_int;
typedef __attribute__((address_space(3))) int l_int;

__device__ __forceinline__ void async_load_index_b32(const int* g, int* l) {
#if HAVE_ASYNC_LDS
  __builtin_amdgcn_global_load_async_to_lds_b32(
      (g_int*)(uintptr_t)g,
      (l_int*)(uint32_t)(uintptr_t)l,
      0, 0);
#else
  *l = *g;
#endif
}

__device__ __forceinline__ void wait_async_then_barrier() {
#if HAVE_ASYNC_LDS
#if __has_builtin(__builtin_amdgcn_s_wait_asynccnt)
  __builtin_amdgcn_s_wait_asynccnt(0);
#else
  asm volatile("s_wait_asynccnt 0" ::: "memory");
#endif
#endif
  __syncthreads();
}

__device__ __forceinline__ float clamp01(float x) {
  return fminf(fmaxf(x, 0.0f), 1.0f);
}

// ---------------- degree ---------------------------------------------------

__global__ __launch_bounds__(256) void lp_zero_u32(unsigned* __restrict__ p, int n) {
  int i = blockIdx.x * 256 + threadIdx.x;
  if (i < n) p[i] = 0u;
}

__global__ __launch_bounds__(256) void lp_degree(const int* __restrict__ dst,
                                                 unsigned* __restrict__ counts,
                                                 int n_edges) {
  int e = blockIdx.x * 256 + threadIdx.x;
  if (e < n_edges) atomicAdd(&counts[dst[e]], 1u);
}

// ---------------- hierarchical exclusive scan (counts -> row_ptr) ---------
// A) per-256-block scan (wave32 shfl + LDS), writes block-local exclusive
//    prefix and per-block totals; also emits norm = rsqrt(max(deg,1)).

__global__ __launch_bounds__(256) void lp_scan_block(const unsigned* __restrict__ counts,
                                                     unsigned* __restrict__ row_ptr,
                                                     unsigned* __restrict__ blk_sums,
                                                     float* __restrict__ norm,
                                                     int n) {
  __shared__ unsigned ws[8];
  const int tid = threadIdx.x;
  const int i = blockIdx.x * 256 + tid;
  unsigned c = (i < n) ? counts[i] : 0u;
  if (i < n) norm[i] = rsqrtf(c ? (float)c : 1.0f);

  const int lane = tid & 31;
  const int wid = tid >> 5;
  unsigned v = c;  // becomes inclusive scan within wave
#pragma unroll
  for (int off = 1; off < 32; off <<= 1) {
    unsigned u = __shfl_up(v, off, 32);
    if (lane >= off) v += u;
  }
  if (lane == 31) ws[wid] = v;
  __syncthreads();
  if (tid == 0) {
    unsigned s = 0;
#pragma unroll
    for (int k = 0; k < 8; ++k) { unsigned x = ws[k]; ws[k] = s; s += x; }
  }
  __syncthreads();
  unsigned excl = ws[wid] + v - c;     // block-local exclusive prefix
  if (i < n) row_ptr[i] = excl;
  if (tid == 255) blk_sums[blockIdx.x] = excl + c;  // block total
}

// B) scan the (<=1024) block totals in a single block.
__global__ __launch_bounds__(1024) void lp_scan_tops(unsigned* __restrict__ blk_sums,
                                                     int nblk) {
  __shared__ unsigned ws[32];
  const int tid = threadIdx.x;
  unsigned c = (tid < nblk) ? blk_sums[tid] : 0u;
  const int lane = tid & 31;
  const int wid = tid >> 5;
  unsigned v = c;
#pragma unroll
  for (int off = 1; off < 32; off <<= 1) {
    unsigned u = __shfl_up(v, off, 32);
    if (lane >= off) v += u;
  }
  if (lane == 31) ws[wid] = v;
  __syncthreads();
  if (tid == 0) {
    unsigned s = 0;
#pragma unroll
    for (int k = 0; k < 32; ++k) { unsigned x = ws[k]; ws[k] = s; s += x; }
  }
  __syncthreads();
  if (tid < nblk) blk_sums[tid] = ws[wid] + v - c;  // exclusive block offsets
}

// C) add block offsets; also init fill cursor and row_ptr[n] = E.
__global__ __launch_bounds__(256) void lp_scan_add(unsigned* __restrict__ row_ptr,
                                                   unsigned* __restrict__ cursor,
                                                   const unsigned* __restrict__ blk_sums,
                                                   int n, int n_edges) {
  int i = blockIdx.x * 256 + threadIdx.x;
  if (i < n) {
    unsigned r = row_ptr[i] + blk_sums[blockIdx.x];
    row_ptr[i] = r;
    cursor[i]  = r;
  }
  if (i == 0) row_ptr[n] = (unsigned)n_edges;
}

// ---------------- CSR fill (edge streams staged via CDNA5 async loads) ----

__global__ __launch_bounds__(256) void lp_fill_csr(const int* __restrict__ src,
                                                   const int* __restrict__ dst,
                                                   unsigned* __restrict__ cursor,
                                                   int* __restrict__ csr_src,
                                                   int n_edges) {
  __shared__ int ls[256];
  __shared__ int ldst[256];
  const int tid = threadIdx.x;
  const long long base = (long long)blockIdx.x * 256;
  if (base + tid < n_edges) {
    async_load_index_b32(src + base + tid, &ls[tid]);
    async_load_index_b32(dst + base + tid, &ldst[tid]);
  }
  wait_async_then_barrier();
  if (base + tid < n_edges) {
    unsigned pos = atomicAdd(&cursor[ldst[tid]], 1u);
    csr_src[pos] = ls[tid];
  }
}

// ---------------- node init: t = norm * mask * labels ---------------------

__global__ __launch_bounds__(256) void lp_init(const float4* __restrict__ labels,
                                               const unsigned char* __restrict__ mask,
                                               const float* __restrict__ norm,
                                               float4* __restrict__ t,
                                               long long n4) {
  long long idx = (long long)blockIdx.x * 256 + threadIdx.x;
  if (idx >= n4) return;
  int node = (int)(idx >> 4);               // 16 float4 per node (64 feats)
  float s = mask[node] ? norm[node] : 0.0f;
  float4 l = labels[idx];
  t[idx] = make_float4(s * l.x, s * l.y, s * l.z, s * l.w);
}

// ---------------- fused gather + update -----------------------------------
// 16 lanes per node, lane owns floats [4*lane .. 4*lane+3].
// acc = sum over incoming edges of t_in[src] (register accumulation, 4-way
// unrolled for ILP across the index->data load chain).
// y = clip(0.1*mask*label + 0.9*norm*acc); emit y (last layer) or norm*y.

__global__ __launch_bounds__(256) void lp_gather_update(
    const unsigned* __restrict__ row_ptr,
    const int* __restrict__ csr_src,
    const float4* __restrict__ t_in,
    const float4* __restrict__ labels,
    const unsigned char* __restrict__ mask,
    const float* __restrict__ norm,
    float4* __restrict__ t_out,
    int emit_y,
    int n_nodes) {
  const int lane = threadIdx.x & 15;
  const int node = blockIdx.x * 16 + (threadIdx.x >> 4);
  if (node >= n_nodes) return;

  const unsigned beg = row_ptr[node];
  const unsigned end = row_ptr[node + 1];

  float4 a0 = make_float4(0.f, 0.f, 0.f, 0.f);
  float4 a1 = make_float4(0.f, 0.f, 0.f, 0.f);
  float4 a2 = make_float4(0.f, 0.f, 0.f, 0.f);
  float4 a3 = make_float4(0.f, 0.f, 0.f, 0.f);

  unsigned e = beg;
  for (; e + 4 <= end; e += 4) {
    int s0 = csr_src[e + 0];
    int s1 = csr_src[e + 1];
    int s2 = csr_src[e + 2];
    int s3 = csr_src[e + 3];
    float4 v0 = t_in[(long long)s0 * 16 + lane];
    float4 v1 = t_in[(long long)s1 * 16 + lane];
    float4 v2 = t_in[(long long)s2 * 16 + lane];
    float4 v3 = t_in[(long long)s3 * 16 + lane];
    a0.x += v0.x; a0.y += v0.y; a0.z += v0.z; a0.w += v0.w;
    a1.x += v1.x; a1.y += v1.y; a1.z += v1.z; a1.w += v1.w;
    a2.x += v2.x; a2.y += v2.y; a2.z += v2.z; a2.w += v2.w;
    a3.x += v3.x; a3.y += v3.y; a3.z += v3.z; a3.w += v3.w;
  }
  for (; e < end; ++e) {
    int s0 = csr_src[e];
    float4 v0 = t_in[(long long)s0 * 16 + lane];
    a0.x += v0.x; a0.y += v0.y; a0.z += v0.z; a0.w += v0.w;
  }
  float4 acc = make_float4((a0.x + a1.x) + (a2.x + a3.x),
                           (a0.y + a1.y) + (a2.y + a3.y),
                           (a0.z + a1.z) + (a2.z + a3.z),
                           (a0.w + a1.w) + (a2.w + a3.w));

  const float nm = norm[node];
  const float lw = mask[node] ? LAST_F : 0.0f;
  const float an = ALPHA_F * nm;
  const long long idx = (long long)node * 16 + lane;
  const float4 l = labels[idx];
  float4 y;
  y.x = clamp01(fmaf(an, acc.x, lw * l.x));
  y.y = clamp01(fmaf(an, acc.y, lw * l.y));
  y.z = clamp01(fmaf(an, acc.z, lw * l.z));
  y.w = clamp01(fmaf(an, acc.w, lw * l.w));

  if (emit_y) {
    t_out[idx] = y;
  } else {
    t_out[idx] = make_float4(nm * y.x, nm * y.y, nm * y.z, nm * y.w);
  }
}

// ---------------- host launch ---------------------------------------------

static inline size_t align256(size_t x) { return (x + 255) & ~(size_t)255; }

extern "C" void kernel_launch(void* const* d_in, const int* in_sizes, int n_in,
                              void* d_out, int out_size, void* d_ws, size_t ws_size,
                              hipStream_t stream) {
  const float4*        labels = (const float4*)d_in[0];
  const int*           src    = (const int*)d_in[1];   // jax demotes int64->int32
  const int*           dst    = (const int*)d_in[2];
  const unsigned char* mask   = (const unsigned char*)d_in[3];

  const int n_nodes = in_sizes[3];              // mask length == N
  const int n_edges = in_sizes[1];
  const long long n4 = (long long)n_nodes * 16; // float4 elements (64 feats)

  // workspace layout
  char* ws = (char*)d_ws;
  size_t off = 0;
  unsigned* counts  = (unsigned*)(ws + off); off += align256((size_t)n_nodes * 4);
  unsigned* row_ptr = (unsigned*)(ws + off); off += align256(((size_t)n_nodes + 1) * 4);
  unsigned* cursor  = (unsigned*)(ws + off); off += align256((size_t)n_nodes * 4);
  unsigned* blksum  = (unsigned*)(ws + off); off += align256(1024 * 4);
  float*    norm    = (float*)(ws + off);    off += align256((size_t)n_nodes * 4);
  int*      csr_src = (int*)(ws + off);      off += align256((size_t)n_edges * 4);
  float4*   tA      = (float4*)(ws + off);   off += (size_t)n4 * sizeof(float4);
  float4*   tB      = (float4*)d_out;        // d_out doubles as ping-pong scratch

  const int gN  = (n_nodes + 255) / 256;   // 391 (<=1024 for the tops scan)
  const int gE  = (n_edges + 255) / 256;   // 12500
  const int g4  = (int)((n4 + 255) / 256); // 6250
  const int gG  = (n_nodes + 15) / 16;     // 6250 (16 nodes per 256-thread block)

  // CSR build (one-time)
  lp_zero_u32<<<gN, 256, 0, stream>>>(counts, n_nodes);
  lp_degree<<<gE, 256, 0, stream>>>(dst, counts, n_edges);
  lp_scan_block<<<gN, 256, 0, stream>>>(counts, row_ptr, blksum, norm, n_nodes);
  lp_scan_tops<<<1, 1024, 0, stream>>>(blksum, gN);
  lp_scan_add<<<gN, 256, 0, stream>>>(row_ptr, cursor, blksum, n_nodes, n_edges);
  lp_fill_csr<<<gE, 256, 0, stream>>>(src, dst, cursor, csr_src, n_edges);

  // propagation
  lp_init<<<g4, 256, 0, stream>>>(labels, mask, norm, tA, n4);
  lp_gather_update<<<gG, 256, 0, stream>>>(row_ptr, csr_src, tA, labels, mask, norm,
                                           tB, 0, n_nodes);            // layer 0
  lp_gather_update<<<gG, 256, 0, stream>>>(row_ptr, csr_src, tB, labels, mask, norm,
                                           tA, 0, n_nodes);            // layer 1
  lp_gather_update<<<gG, 256, 0, stream>>>(row_ptr, csr_src, tA, labels, mask, norm,
                                           (float4*)d_out, 1, n_nodes); // layer 2 -> y
}